// InferencePipeline_57612691309297
// MI455X (gfx1250) — compile-verified
//
#include <hip/hip_runtime.h>
#include <hip/hip_bf16.h>
#include <math.h>

typedef __attribute__((ext_vector_type(2))) float v2f;
typedef __attribute__((ext_vector_type(8))) float v8f;

#define CCH   80
#define PH    128
#define PW    256
#define PHW   (PH * PW)        // 32768 (thing_map / kernel_space spatial)
#define NTOT  (CCH * PHW)      // 2621440
#define DDIM  256              // kernel dim
#define EHW   131072           // embeddings spatial 256*512
#define KSEL  50               // top-k
#define MPAD  64               // padded M for WMMA (4 tiles of 16)
#define NB1   2048             // argmax chunk count (stage-1 blocks)
#define SPAN  (NB1 * 256)      // 524288; NTOT = 5*SPAN exactly

// fast sigmoid: native v_exp_f32 + v_rcp_f32 (avoids IEEE div chain in epilogue)
__device__ __forceinline__ float sigmoidf_(float x) {
    float e = __expf(-x);
    return __builtin_amdgcn_rcpf(1.0f + e);
}

// ---------------- peak detection ----------------
__global__ void k_sigmoid(const float* __restrict__ x, float* __restrict__ S) {
    int i = blockIdx.x * blockDim.x + threadIdx.x;
    if (i < NTOT) S[i] = sigmoidf_(x[i]);
}

// centers = 0.5*(sig + avg3x3(sig));  avg divides by 9 (count_include_pad semantics)
__global__ void k_combine(const float* __restrict__ S, float* __restrict__ Cb) {
    int i = blockIdx.x * blockDim.x + threadIdx.x;
    if (i >= NTOT) return;
    int c = i / PHW, r = i - c * PHW, h = r / PW, w = r - h * PW;
    const float* Sc = S + c * PHW;
    float s = 0.0f;
    for (int dh = -1; dh <= 1; ++dh) {
        int hh = h + dh; if (hh < 0 || hh >= PH) continue;
        for (int dw = -1; dw <= 1; ++dw) {
            int ww = w + dw; if (ww < 0 || ww >= PW) continue;
            s += Sc[hh * PW + ww];
        }
    }
    Cb[i] = 0.5f * (Sc[h * PW + w] + s * (1.0f / 9.0f));
}

// P = Cb * (max3x3(Cb) == Cb)
__global__ void k_peak(const float* __restrict__ Cb, float* __restrict__ P) {
    int i = blockIdx.x * blockDim.x + threadIdx.x;
    if (i >= NTOT) return;
    int c = i / PHW, r = i - c * PHW, h = r / PW, w = r - h * PW;
    const float* Cc = Cb + c * PHW;
    float v = Cc[h * PW + w];
    float m = -3.0e38f;
    for (int dh = -1; dh <= 1; ++dh) {
        int hh = h + dh; if (hh < 0 || hh >= PH) continue;
        for (int dw = -1; dw <= 1; ++dw) {
            int ww = w + dw; if (ww < 0 || ww >= PW) continue;
            float t = Cc[hh * PW + ww];
            m = fmaxf(m, t);
        }
    }
    P[i] = (m == v) ? v : 0.0f;
}

// ---------------- top-k: per-chunk argmax candidates (one pass) ----------------
__global__ void k_argmax1(const float* __restrict__ P,
                          float* __restrict__ cv, int* __restrict__ ci) {
    __shared__ float sv[256];
    __shared__ int   si[256];
    int t = threadIdx.x;
    float bv = -3.0f; int bi = 0x7fffffff;
    for (int i = blockIdx.x * 256 + t; i < NTOT; i += SPAN) {
        float v = P[i];
        if (v > bv) { bv = v; bi = i; }   // increasing i -> lowest index wins ties
    }
    sv[t] = bv; si[t] = bi;
    __syncthreads();
    for (int off = 128; off > 0; off >>= 1) {
        if (t < off) {
            float v2 = sv[t + off]; int i2 = si[t + off];
            if (v2 > sv[t] || (v2 == sv[t] && i2 < si[t])) { sv[t] = v2; si[t] = i2; }
        }
        __syncthreads();
    }
    if (t == 0) { cv[blockIdx.x] = sv[0]; ci[blockIdx.x] = si[0]; }
}

// ---------------- top-k: 50 rounds in ONE kernel (single block) ----------------
// Each round: reduce 2048 chunk candidates -> winner; mark P[idx]=-1; rescan only
// the winner's chunk (its exact original element set) to refresh its candidate.
__global__ void k_top50(float* __restrict__ P,
                        float* __restrict__ cv, int* __restrict__ ci,
                        float* __restrict__ selv, int* __restrict__ seli) {
    __shared__ float sv[256];
    __shared__ int   si[256];
    __shared__ int   schunk;
    int t = threadIdx.x;
    for (int k = 0; k < KSEL; ++k) {
        // 1) global winner from chunk candidates
        float bv = -3.0f; int bi = 0x7fffffff;
        for (int i = t; i < NB1; i += 256) {
            float v = cv[i]; int ii = ci[i];
            if (v > bv || (v == bv && ii < bi)) { bv = v; bi = ii; }
        }
        sv[t] = bv; si[t] = bi;
        __syncthreads();
        for (int off = 128; off > 0; off >>= 1) {
            if (t < off) {
                float v2 = sv[t + off]; int i2 = si[t + off];
                if (v2 > sv[t] || (v2 == sv[t] && i2 < si[t])) { sv[t] = v2; si[t] = i2; }
            }
            __syncthreads();
        }
        if (t == 0) {
            int idx = si[0];
            selv[k] = sv[0];
            seli[k] = idx;
            P[idx]  = -1.0f;                 // exclude (real values are >= 0)
            schunk  = (idx % SPAN) / 256;    // chunk that owned the winner
            __threadfence_block();
        }
        __syncthreads();
        // 2) refresh that chunk's candidate
        int c = schunk;
        bv = -3.0f; bi = 0x7fffffff;
        for (int i = c * 256 + t; i < NTOT; i += SPAN) {
            float v = P[i];
            if (v > bv) { bv = v; bi = i; }
        }
        sv[t] = bv; si[t] = bi;
        __syncthreads();
        for (int off = 128; off > 0; off >>= 1) {
            if (t < off) {
                float v2 = sv[t + off]; int i2 = si[t + off];
                if (v2 > sv[t] || (v2 == sv[t] && i2 < si[t])) { sv[t] = v2; si[t] = i2; }
            }
            __syncthreads();
        }
        if (t == 0) { cv[c] = sv[0]; ci[c] = si[0]; }
        __syncthreads();
    }
}

// ---------------- gather kernels -> padded A [64 x 256] ----------------
__global__ void k_gather(const float* __restrict__ ksp, const int* __restrict__ seli,
                         float* __restrict__ A) {
    int row = blockIdx.x;       // 0..63
    int d   = threadIdx.x;      // 0..255
    float v = 0.0f;
    if (row < KSEL) {
        int spatial = seli[row] % PHW;
        v = ksp[d * PHW + spatial];
    }
    A[row * DDIM + d] = v;
}

__global__ void k_init(float* __restrict__ sums, float* __restrict__ cnts) {
    int t = threadIdx.x;
    if (t < MPAD) { sums[t] = 0.0f; cnts[t] = 0.0f; }
}

// ---------------- WMMA GEMM + sigmoid + rescoring partials ----------------
// masks[k, p] = sigmoid( sum_d A[k,d] * E[d,p] ),  k<50, p<131072
// grid.x = EHW/128 ; block = 256 (8 waves). Wave w owns columns [blk*128+w*16, +16),
// holds 4 accumulators (M-tiles 0..3) so each E element is read exactly once per block.
__global__ void k_gemm(const float* __restrict__ A, const float* __restrict__ E,
                       float* __restrict__ masks,
                       float* __restrict__ sums, float* __restrict__ cnts) {
    __shared__ float lsum[MPAD];
    __shared__ float lcnt[MPAD];
    int t = threadIdx.x;
    if (t < MPAD) { lsum[t] = 0.0f; lcnt[t] = 0.0f; }
    __syncthreads();

    int wave = t >> 5;
    int lane = t & 31;
    int col  = blockIdx.x * 128 + wave * 16 + (lane & 15);
    int klo  = (lane < 16) ? 0 : 2;     // A/B fragment K-halves per ISA layout
    int arow = lane & 15;

    v8f acc[4];
    acc[0] = (v8f)0.0f; acc[1] = (v8f)0.0f; acc[2] = (v8f)0.0f; acc[3] = (v8f)0.0f;

    for (int kk = 0; kk < DDIM; kk += 4) {
        v2f b;
        b.x = E[(size_t)(kk + klo)     * EHW + col];
        b.y = E[(size_t)(kk + klo + 1) * EHW + col];
#pragma unroll
        for (int mt = 0; mt < 4; ++mt) {
            const float* Ar = A + (mt * 16 + arow) * DDIM + kk + klo;
            v2f a;
            a.x = Ar[0];
            a.y = Ar[1];
            acc[mt] = __builtin_amdgcn_wmma_f32_16x16x4_f32(
                false, a, false, b, (short)0, acc[mt], false, false);
        }
    }

    // C/D layout: VGPR r, lanes 0-15 -> M=r ; lanes 16-31 -> M=r+8 ; N = lane&15
    int roff = (lane < 16) ? 0 : 8;
#pragma unroll
    for (int mt = 0; mt < 4; ++mt) {
#pragma unroll
        for (int r = 0; r < 8; ++r) {
            int grow = mt * 16 + roff + r;
            if (grow < KSEL) {
                float m = sigmoidf_(acc[mt][r]);
                masks[(size_t)grow * EHW + col] = m;
                if (m > 0.4f) {
                    atomicAdd(&lsum[grow], m);
                    atomicAdd(&lcnt[grow], 1.0f);
                }
            }
        }
    }
    __syncthreads();
    if (t < MPAD) {
        if (lsum[t] != 0.0f) atomicAdd(&sums[t], lsum[t]);
        if (lcnt[t] != 0.0f) atomicAdd(&cnts[t], lcnt[t]);
    }
}

// ---------------- finalize scores / cats ----------------
__global__ void k_final(const float* __restrict__ selv, const int* __restrict__ seli,
                        const float* __restrict__ sums, const float* __restrict__ cnts,
                        float* __restrict__ oscores, float* __restrict__ ocats) {
    int k = threadIdx.x;
    if (k < KSEL) {
        float sf = sums[k] / fmaxf(cnts[k], 1e-8f);
        float v  = selv[k];
        float valid = (v > 0.1f) ? 1.0f : 0.0f;
        oscores[k] = v * sf * valid;
        ocats[k]   = (float)(seli[k] / PHW);
    }
}

extern "C" void kernel_launch(void* const* d_in, const int* in_sizes, int n_in,
                              void* d_out, int out_size, void* d_ws, size_t ws_size,
                              hipStream_t stream) {
    const float* thing = (const float*)d_in[0];   // [1,80,128,256]
    const float* ksp   = (const float*)d_in[1];   // [1,256,128,256]
    const float* emb   = (const float*)d_in[2];   // [1,256,256,512]

    float* out = (float*)d_out;
    // d_out scratch reuse (overwritten by GEMM later, after top-k completes):
    float* S  = out;          // NTOT floats: sigmoid map, then peak map P
    float* Cb = out + NTOT;   // NTOT floats: combined centers
    float* masks      = out;                      // [50, 131072]
    float* out_scores = out + (size_t)KSEL * EHW; // [50]
    float* out_cats   = out_scores + KSEL;        // [50]

    char* ws = (char*)d_ws;
    float* cv   = (float*)(ws);                            // 2048
    int*   ci   = (int*)  (ws + NB1 * 4);                  // 2048
    float* selv = (float*)(ws + NB1 * 8);                  // 64
    int*   seli = (int*)  (ws + NB1 * 8 + 256);            // 64
    float* A    = (float*)(ws + NB1 * 8 + 512);            // 64*256
    float* sums = (float*)(ws + NB1 * 8 + 512 + MPAD * DDIM * 4); // 64
    float* cnts = sums + MPAD;                                    // 64

    dim3 b256(256);
    int gb = (NTOT + 255) / 256;

    k_sigmoid<<<gb, b256, 0, stream>>>(thing, S);
    k_combine<<<gb, b256, 0, stream>>>(S, Cb);
    k_peak   <<<gb, b256, 0, stream>>>(Cb, S);     // peak map written over S

    k_init<<<1, 128, 0, stream>>>(sums, cnts);

    k_argmax1<<<NB1, b256, 0, stream>>>(S, cv, ci);
    k_top50  <<<1,   b256, 0, stream>>>(S, cv, ci, selv, seli);

    k_gather<<<MPAD, b256, 0, stream>>>(ksp, seli, A);
    k_gemm  <<<EHW / 128, b256, 0, stream>>>(A, emb, masks, sums, cnts);
    k_final <<<1, 64, 0, stream>>>(selv, seli, sums, cnts, out_scores, out_cats);
}